// QuantumAutoencoder_86792699118142
// MI455X (gfx1250) — compile-verified
//
#include <hip/hip_runtime.h>
#include <math.h>

// ---------------------------------------------------------------------------
// Quantum autoencoder: 12-qubit circuit sim (one WGP, state in LDS, gate
// application via V_WMMA_F32_16X16X4_F32) + tiny MLP, then a bandwidth-bound
// broadcast of the 16 resulting pixels to (B,1,4,4) = 64 MB.
// ---------------------------------------------------------------------------

typedef float v2f __attribute__((ext_vector_type(2)));
typedef float v8f __attribute__((ext_vector_type(8)));

#define NQ 12
#define NL 3
#define DIM 4096          // 2^12 amplitudes
#define THREADS 512       // 16 wave32 waves -> 16 column tiles of 16

struct cpx { float re, im; };
__device__ inline cpx cmul(cpx a, cpx b) {
    return { a.re * b.re - a.im * b.im, a.re * b.im + a.im * b.re };
}
__device__ inline cpx cadd(cpx a, cpx b) { return { a.re + b.re, a.im + b.im }; }

__device__ inline v8f wmma_f32(v2f a, v2f b, v8f c) {
    // D(16x16) += A(16x4) * B(4x16), fp32 (V_WMMA_F32_16X16X4_F32)
    return __builtin_amdgcn_wmma_f32_16x16x4_f32(
        /*neg_a=*/false, a, /*neg_b=*/false, b,
        /*c_mod=*/(short)0, c, /*reuse_a=*/false, /*reuse_b=*/false);
}

// Permuted flat address of (group-axis row r, column col) for group g.
// State flat index = a*256 + b*16 + c  (a = qubits 0-3, b = 4-7, c = 8-11).
__device__ inline int gaddr(int g, int r, int col) {
    if (g == 0) return r * 256 + col;                               // rows = a, cols = (b,c)
    if (g == 1) return (col >> 4) * 256 + r * 16 + (col & 15);      // rows = b, cols = (a,c)
    return (col >> 4) * 256 + (col & 15) * 16 + r;                  // rows = c, cols = (a,b)
}

__global__ __launch_bounds__(THREADS, 1)
void qae_sim_kernel(const float* __restrict__ qw,
                    const float* __restrict__ W1, const float* __restrict__ b1,
                    const float* __restrict__ W2, const float* __restrict__ b2,
                    float* __restrict__ pix_out)
{
    __shared__ float Sre[DIM], Sim[DIM];      // state (32 KB)
    __shared__ float Ure[256], Uim[256];      // current 16x16 group unitary
    __shared__ cpx   G[NQ][2][2];             // per-qubit gates for this layer
    __shared__ float wsum[16][NQ];            // per-wave partial <Z_q>
    __shared__ float zv[NQ];
    __shared__ float hv[32];

    const int tid  = threadIdx.x;
    const int lane = tid & 31;
    const int wave = tid >> 5;
    const int half = (lane >> 4) & 1;   // lane half selects K / row offset
    const int ln   = lane & 15;

    // |0...0>
    for (int k = tid; k < DIM; k += THREADS) { Sre[k] = 0.f; Sim[k] = 0.f; }
    if (tid == 0) Sre[0] = 1.f;
    __syncthreads();

    for (int l = 0; l < NL; ++l) {
        // ---- per-qubit gate matrices  G = RZ * RY * RX ----
        if (tid < NQ) {
            float tx = qw[(l * NQ + tid) * 3 + 0];
            float ty = qw[(l * NQ + tid) * 3 + 1];
            float tz = qw[(l * NQ + tid) * 3 + 2];
            float cx = cosf(0.5f * tx), sx = sinf(0.5f * tx);
            float cy = cosf(0.5f * ty), sy = sinf(0.5f * ty);
            float cz = cosf(0.5f * tz), sz = sinf(0.5f * tz);
            cpx RX[2][2] = { { {cx, 0.f}, {0.f, -sx} }, { {0.f, -sx}, {cx, 0.f} } };
            cpx RY[2][2] = { { {cy, 0.f}, {-sy, 0.f} }, { {sy, 0.f},  {cy, 0.f} } };
            cpx RZ[2][2] = { { {cz, -sz}, {0.f, 0.f} }, { {0.f, 0.f}, {cz, sz}  } };
            cpx M[2][2];
            for (int r = 0; r < 2; ++r)
                for (int c = 0; c < 2; ++c)
                    M[r][c] = cadd(cmul(RY[r][0], RX[0][c]), cmul(RY[r][1], RX[1][c]));
            for (int r = 0; r < 2; ++r)
                for (int c = 0; c < 2; ++c)
                    G[tid][r][c] = cadd(cmul(RZ[r][0], M[0][c]), cmul(RZ[r][1], M[1][c]));
        }
        __syncthreads();

        // ---- three 4-qubit group unitaries, applied with f32 WMMA ----
        for (int g = 0; g < 3; ++g) {
            // U = kron(G[4g], G[4g+1], G[4g+2], G[4g+3]); qubit 4g acts on MSB
            if (tid < 256) {
                int r = tid >> 4, c = tid & 15;
                cpx v = { 1.f, 0.f };
                for (int k = 0; k < 4; ++k) {
                    int rb = (r >> (3 - k)) & 1, cb = (c >> (3 - k)) & 1;
                    v = cmul(v, G[g * 4 + k][rb][cb]);
                }
                Ure[tid] = v.re; Uim[tid] = v.im;
            }
            __syncthreads();

            const int col = 16 * wave + ln;   // this wave's column (N index)

            // B fragments: chunk kk, vgpr v -> row k = 4kk + v + 2*half, this col
            float bre[4][2], bim[4][2];
            #pragma unroll
            for (int kk = 0; kk < 4; ++kk)
                #pragma unroll
                for (int v = 0; v < 2; ++v) {
                    int a = gaddr(g, 4 * kk + v + 2 * half, col);
                    bre[kk][v] = Sre[a];
                    bim[kk][v] = Sim[a];
                }
            // A fragments from U: row m = ln, k = 4kk + v + 2*half
            float are[4][2], aim[4][2];
            #pragma unroll
            for (int kk = 0; kk < 4; ++kk)
                #pragma unroll
                for (int v = 0; v < 2; ++v) {
                    int kidx = 4 * kk + v + 2 * half;
                    are[kk][v] = Ure[ln * 16 + kidx];
                    aim[kk][v] = Uim[ln * 16 + kidx];
                }

            v8f accre, accim;
            #pragma unroll
            for (int i = 0; i < 8; ++i) { accre[i] = 0.f; accim[i] = 0.f; }

            // Dre = Ure*Sre + Uim*(-Sim);  Dim = Ure*Sim + Uim*Sre
            #pragma unroll
            for (int kk = 0; kk < 4; ++kk) {
                v2f Ar;  Ar[0]  = are[kk][0];  Ar[1]  = are[kk][1];
                v2f Ai;  Ai[0]  = aim[kk][0];  Ai[1]  = aim[kk][1];
                v2f Br;  Br[0]  = bre[kk][0];  Br[1]  = bre[kk][1];
                v2f Bi;  Bi[0]  = bim[kk][0];  Bi[1]  = bim[kk][1];
                v2f Bin; Bin[0] = -bim[kk][0]; Bin[1] = -bim[kk][1];
                accre = wmma_f32(Ar, Br,  accre);
                accre = wmma_f32(Ai, Bin, accre);
                accim = wmma_f32(Ar, Bi,  accim);
                accim = wmma_f32(Ai, Br,  accim);
            }

            // D store: vgpr v -> row m = v + 8*half, this col (in-place: tile-disjoint)
            #pragma unroll
            for (int v = 0; v < 8; ++v) {
                int a = gaddr(g, v + 8 * half, col);
                Sre[a] = accre[v];
                Sim[a] = accim[v];
            }
            __syncthreads();
        }

        // ---- CNOT ring, sequential (adjacent CNOTs don't commute) ----
        for (int q = 0; q < NQ; ++q) {
            int pc = 11 - q;                    // control bit position
            int pt = 11 - ((q + 1) % NQ);       // target bit position
            int hi = pc > pt ? pc : pt, lo = pc > pt ? pt : pc;
            for (int pp = tid; pp < 1024; pp += THREADS) {
                unsigned u = (unsigned)pp;
                unsigned lowm = u & ((1u << lo) - 1u); u >>= lo;
                unsigned midm = u & ((1u << (hi - lo - 1)) - 1u); u >>= (hi - lo - 1);
                unsigned i = (u << (hi + 1)) | (midm << (lo + 1)) | lowm;
                i |= (1u << pc);                // control = 1, target currently 0
                unsigned j = i | (1u << pt);
                float tr = Sre[i], ti = Sim[i];
                Sre[i] = Sre[j]; Sim[i] = Sim[j];
                Sre[j] = tr;     Sim[j] = ti;
            }
            __syncthreads();
        }
    }

    // ---- <Z_q> expectations (deterministic shuffle reduction) ----
    float ez[NQ];
    #pragma unroll
    for (int q = 0; q < NQ; ++q) ez[q] = 0.f;
    for (int k = tid; k < DIM; k += THREADS) {
        float p = Sre[k] * Sre[k] + Sim[k] * Sim[k];
        #pragma unroll
        for (int q = 0; q < NQ; ++q)
            ez[q] += ((k >> (11 - q)) & 1) ? -p : p;
    }
    #pragma unroll
    for (int q = 0; q < NQ; ++q)
        for (int off = 16; off >= 1; off >>= 1)
            ez[q] += __shfl_xor(ez[q], off, 32);
    if (lane == 0)
        for (int q = 0; q < NQ; ++q) wsum[wave][q] = ez[q];
    __syncthreads();
    if (tid < NQ) {
        float s = 0.f;
        for (int w = 0; w < 16; ++w) s += wsum[w][tid];
        zv[tid] = s;
    }
    __syncthreads();

    // ---- MLP 12 -> 32 (relu) -> 16 (sigmoid) ----
    if (tid < 32) {
        float s = b1[tid];
        for (int k = 0; k < NQ; ++k) s += zv[k] * W1[tid * NQ + k];
        hv[tid] = fmaxf(s, 0.f);
    }
    __syncthreads();
    if (tid < 16) {
        float s = b2[tid];
        for (int j = 0; j < 32; ++j) s += hv[j] * W2[tid * 32 + j];
        pix_out[tid] = 1.f / (1.f + expf(-s));
    }
}

// ---------------------------------------------------------------------------
// Broadcast 16 floats to B*16 output floats (store-bandwidth bound)
// ---------------------------------------------------------------------------
__global__ __launch_bounds__(256)
void qae_bcast_kernel(const float* __restrict__ pix, float4* __restrict__ out, int n4)
{
    __shared__ float4 p4[4];
    if (threadIdx.x < 4) p4[threadIdx.x] = ((const float4*)pix)[threadIdx.x];
    __syncthreads();
    int gid = blockIdx.x * blockDim.x + threadIdx.x;
    if (gid < n4) out[gid] = p4[gid & 3];
}

extern "C" void kernel_launch(void* const* d_in, const int* in_sizes, int n_in,
                              void* d_out, int out_size, void* d_ws, size_t ws_size,
                              hipStream_t stream)
{
    // setup_inputs order: images(unused), qweights, W1, b1, W2, b2
    const float* qw = (const float*)d_in[1];
    const float* W1 = (const float*)d_in[2];
    const float* b1 = (const float*)d_in[3];
    const float* W2 = (const float*)d_in[4];
    const float* b2 = (const float*)d_in[5];
    float* pix = (float*)d_ws;   // 16 floats of scratch

    qae_sim_kernel<<<1, THREADS, 0, stream>>>(qw, W1, b1, W2, b2, pix);

    int n4 = out_size / 4;                    // float4 stores
    int blocks = (n4 + 255) / 256;
    qae_bcast_kernel<<<blocks, 256, 0, stream>>>(pix, (float4*)d_out, n4);
}